// MultiHeadAttention_74491912782210
// MI455X (gfx1250) — compile-verified
//
#include <hip/hip_runtime.h>
#include <hip/hip_bf16.h>

typedef __attribute__((ext_vector_type(2))) float v2f;
typedef __attribute__((ext_vector_type(8))) float v8f;

#define MODEL   1024
#define HEADS   16
#define DH      64
#define BATCH   64
#define NSPLIT  4
#define SUB     16            // tokens per pipelined sub-chunk
#define ROWP    1028          // padded LDS row stride (floats): conflict-free WMMA fetches
#define LN_EPS  1e-5f

// workspace layout (float offsets)
#define OFF_QK   0
#define SZ_QK    (BATCH*HEADS*MODEL)
#define OFF_M    (OFF_QK + SZ_QK)
#define SZ_M     (BATCH*NSPLIT*HEADS)
#define OFF_L    (OFF_M + SZ_M)
#define OFF_CTXP (OFF_L + SZ_M)
#define SZ_CTXP  (BATCH*NSPLIT*HEADS*MODEL)
#define OFF_CTX  (OFF_CTXP + SZ_CTXP)

// ---------------------------------------------------------------------------
// K1: q = x[b, index, :] @ Wq  ;  qk[b,h,m] = (1/8) * sum_d Wk[m, h*64+d] * q[h*64+d]
// ---------------------------------------------------------------------------
__global__ __launch_bounds__(256) void qk_kernel(const float* __restrict__ X,
    const float* __restrict__ Wq, const float* __restrict__ Wk,
    const int* __restrict__ dIndex, float* __restrict__ ws, int Smax)
{
    __shared__ float xl[MODEL];
    __shared__ float ql[MODEL];
    const int b = blockIdx.x, tid = threadIdx.x;
    const int idxv = *dIndex;
    const float* xrow = X + ((size_t)b * Smax + idxv) * MODEL;
    for (int q = 0; q < 4; q++) xl[q*256 + tid] = xrow[q*256 + tid];
    __syncthreads();
    for (int q = 0; q < 4; q++) {
        int c = q*256 + tid;
        float s = 0.f;
        for (int m = 0; m < MODEL; m++) s += xl[m] * Wq[(size_t)m*MODEL + c];
        ql[c] = s * 0.125f;              // fold 1/sqrt(D_K)
    }
    __syncthreads();
    float* qk = ws + OFF_QK + (size_t)b * HEADS * MODEL;
    for (int h = 0; h < HEADS; h++) {
        const float* qh = ql + h*DH;
        for (int q = 0; q < 4; q++) {
            int m = q*256 + tid;
            const float* wr = Wk + (size_t)m*MODEL + h*DH;
            float s = 0.f;
            for (int d = 0; d < DH; d++) s += wr[d] * qh[d];
            qk[h*MODEL + m] = s;
        }
    }
}

// ---------------------------------------------------------------------------
// K2: flash-decoding over one S-split per block. fp32 WMMA for scores and ctx.
// Register-staged software pipeline: while WMMAs chew on the LDS-resident
// sub-chunk, the next sub-chunk's global loads are already in flight (their
// only consumer is next iteration's ds_store, so no wait blocks the math).
// ---------------------------------------------------------------------------
__global__ __launch_bounds__(256) void attn_kernel(const float* __restrict__ X,
    const int* __restrict__ dIndex, float* __restrict__ ws, int Smax)
{
    extern __shared__ float smem[];
    float* xs    = smem;                    // [SUB][ROWP]   X sub-chunk
    float* qks   = xs  + SUB*ROWP;          // [HEADS][ROWP] qk vectors
    float* scr   = qks + HEADS*ROWP;        // [8][256]      K-split score partials
    float* sc    = scr + 8*256;             // [SUB][HEADS]  scores
    float* Pm    = sc  + SUB*HEADS;         // [SUB][HEADS]  probs
    float* scale = Pm  + SUB*HEADS;         // [HEADS]       running-max rescale

    const int tid  = threadIdx.x;
    const int wave = tid >> 5, lane = tid & 31;
    const int b = blockIdx.x / NSPLIT, split = blockIdx.x % NSPLIT;
    const int chunk = Smax / NSPLIT;
    const int nsub  = chunk / SUB;
    const int idxv  = *dIndex;

    const bool laneLow = lane < 16;
    const int  lm = laneLow ? lane : lane - 16;
    const int  c0 = laneLow ? 0 : 2;

    // stage qk[b] into padded LDS
    {
        const float* qk = ws + OFF_QK + (size_t)b * HEADS * MODEL;
        for (int q = 0; q < 16; q++) {
            int lin = q*256 + tid;
            int h = lin >> 8, d4 = (lin & 255) * 4;
            *(float4*)(qks + h*ROWP + d4) = *(const float4*)(qk + h*MODEL + d4);
        }
    }

    float mrun = -3e38f, lrun = 0.f;        // live in threads 0..15 (per head)
    v8f acc[8];                             // ctx tiles: M=head, N=16 dims each
    for (int i = 0; i < 8; i++)
        for (int r = 0; r < 8; r++) acc[i][r] = 0.f;

    const float* xbase = X + (size_t)b * Smax * MODEL;

    // prologue: stage sub-chunk 0 into registers
    float4 stg[16];
    {
        const float* xg = xbase + (size_t)(split*chunk) * MODEL;
        for (int q = 0; q < 16; q++) {
            int lin = q*256 + tid;
            int tok = lin >> 8, d4 = (lin & 255) * 4;
            stg[q] = *(const float4*)(xg + (size_t)tok*MODEL + d4);
        }
    }
    __syncthreads();

    for (int j = 0; j < nsub; j++) {
        const int t0 = split*chunk + j*SUB;

        // ---- commit staged registers -> LDS ----
        for (int q = 0; q < 16; q++) {
            int lin = q*256 + tid;
            int tok = lin >> 8, d4 = (lin & 255) * 4;
            *(float4*)(xs + tok*ROWP + d4) = stg[q];
        }
        __syncthreads();

        // ---- kick off next sub-chunk's global loads (overlaps with WMMAs) ----
        {
            const int tn = (j + 1 < nsub) ? t0 + SUB : t0;   // last iter: benign reload
            const float* xg = xbase + (size_t)tn * MODEL;
            for (int q = 0; q < 16; q++) {
                int lin = q*256 + tid;
                int tok = lin >> 8, d4 = (lin & 255) * 4;
                stg[q] = *(const float4*)(xg + (size_t)tok*MODEL + d4);
            }
        }

        // ---- scores: X(16x1024) @ qk^T(1024x16), K split over the 8 waves ----
        {
            v8f s;
            for (int r = 0; r < 8; r++) s[r] = 0.f;
            const float* xr = xs  + lm * ROWP;                // A: M=token
            const float* qr = qks + lm * ROWP;                // B: N=head
            for (int k4 = 0; k4 < 32; k4++) {
                int dim = wave*128 + k4*4 + c0;
                v2f a  = *(const v2f*)(xr + dim);
                v2f bb = *(const v2f*)(qr + dim);
                s = __builtin_amdgcn_wmma_f32_16x16x4_f32(false, a, false, bb,
                                                          (short)0, s, false, false);
            }
            float* so = scr + wave * 256;
            for (int r = 0; r < 8; r++) so[r*32 + lane] = s[r];
        }
        __syncthreads();

        // ---- reduce K-split partials -> sc[token][head], apply index mask ----
        {
            int pos = tid;
            float sum = 0.f;
            for (int p = 0; p < 8; p++) sum += scr[p*256 + pos];
            int r = pos >> 5, l = pos & 31;
            int tok  = r + ((l >= 16) ? 8 : 0);
            int head = l & 15;
            if (t0 + tok > idxv) sum = -3e38f;
            sc[tok*HEADS + head] = sum;
        }
        __syncthreads();

        // ---- online softmax (one thread per head) ----
        if (tid < 16) {
            const int h = tid;
            float cmax = -3e38f;
            for (int t = 0; t < SUB; t++) cmax = fmaxf(cmax, sc[t*HEADS + h]);
            float mnew = fmaxf(mrun, cmax);
            float scl  = __expf(mrun - mnew);
            float sum = 0.f;
            for (int t = 0; t < SUB; t++) {
                float sv = sc[t*HEADS + h];
                float p = (sv < -1e37f) ? 0.f : __expf(sv - mnew);
                Pm[t*HEADS + h] = p;
                sum += p;
            }
            lrun = lrun * scl + sum;
            mrun = mnew;
            scale[h] = scl;
        }
        __syncthreads();

        // ---- rescale ctx accumulators, then ctx += P^T(16x16) @ X(16x1024) ----
        {
            float sr[8];
            for (int r = 0; r < 8; r++) sr[r] = scale[r + (laneLow ? 0 : 8)];
            for (int i = 0; i < 8; i++)
                for (int r = 0; r < 8; r++) acc[i][r] *= sr[r];

            const int nb = wave * 128;                 // this wave's dim base
            for (int kk = 0; kk < 4; kk++) {
                v2f a;                                  // A: M=head, K=token
                a.x = Pm[(4*kk + (laneLow ? 0 : 2)) * HEADS + lm];
                a.y = Pm[(4*kk + (laneLow ? 1 : 3)) * HEADS + lm];
                const float* x0 = xs + (4*kk + (laneLow ? 0 : 2)) * ROWP;
                const float* x1 = xs + (4*kk + (laneLow ? 1 : 3)) * ROWP;
                for (int i = 0; i < 8; i++) {
                    int n0 = nb + i*16 + lm;
                    v2f bb; bb.x = x0[n0]; bb.y = x1[n0];  // B: K=token, N=dim
                    acc[i] = __builtin_amdgcn_wmma_f32_16x16x4_f32(false, a, false, bb,
                                                                   (short)0, acc[i], false, false);
                }
            }
        }
        __syncthreads();
    }

    // ---- write per-split partials ----
    const int pb = (b*NSPLIT + split) * HEADS;
    if (tid < 16) {
        ws[OFF_M + pb + tid] = mrun;
        ws[OFF_L + pb + tid] = lrun;
    }
    float* ctxp = ws + OFF_CTXP + (size_t)pb * MODEL;
    for (int i = 0; i < 8; i++) {
        int n0 = wave*128 + i*16 + (lane & 15);
        for (int r = 0; r < 8; r++) {
            int head = r + (laneLow ? 0 : 8);
            ctxp[(size_t)head*MODEL + n0] = acc[i][r];
        }
    }
}

// ---------------------------------------------------------------------------
// K3: combine split partials -> ctx[b,h,:]
// ---------------------------------------------------------------------------
__global__ __launch_bounds__(256) void combine_kernel(float* __restrict__ ws)
{
    __shared__ float coef[NSPLIT];
    const int bh = blockIdx.x, b = bh >> 4, h = bh & 15, tid = threadIdx.x;
    if (tid == 0) {
        float mstar = -3e38f;
        for (int s = 0; s < NSPLIT; s++)
            mstar = fmaxf(mstar, ws[OFF_M + (b*NSPLIT + s)*HEADS + h]);
        float denom = 0.f;
        for (int s = 0; s < NSPLIT; s++) {
            float w = __expf(ws[OFF_M + (b*NSPLIT + s)*HEADS + h] - mstar);
            coef[s] = w;
            denom += w * ws[OFF_L + (b*NSPLIT + s)*HEADS + h];
        }
        float inv = 1.f / denom;
        for (int s = 0; s < NSPLIT; s++) coef[s] *= inv;
    }
    __syncthreads();
    const int d0 = tid * 4;
    float4 acc = {0.f, 0.f, 0.f, 0.f};
    for (int s = 0; s < NSPLIT; s++) {
        float4 v = *(const float4*)(ws + OFF_CTXP
                     + ((size_t)((b*NSPLIT + s)*HEADS + h)) * MODEL + d0);
        float c = coef[s];
        acc.x += c*v.x; acc.y += c*v.y; acc.z += c*v.z; acc.w += c*v.w;
    }
    *(float4*)(ws + OFF_CTX + ((size_t)(b*HEADS + h)) * MODEL + d0) = acc;
}

// ---------------------------------------------------------------------------
// K4: head_out = ctx @ Wv ; integrated = head_out @ fcW^T + bias ; LayerNorm
// ---------------------------------------------------------------------------
__global__ __launch_bounds__(256) void out_kernel(const float* __restrict__ Wv,
    const float* __restrict__ fcW, const float* __restrict__ fcB,
    const float* __restrict__ gamma, const float* __restrict__ beta,
    const float* __restrict__ ctxg, float* __restrict__ out)
{
    __shared__ float ho[MODEL];
    __shared__ float redS[256], redQ[256];
    const int b = blockIdx.x, tid = threadIdx.x;
    const float* ctx = ctxg + (size_t)b * HEADS * MODEL;
    for (int q = 0; q < 4; q++) {
        int jj = q*256 + tid;
        const float* cr = ctx + (jj >> 6) * MODEL;
        float s = 0.f;
        for (int m = 0; m < MODEL; m++) s += cr[m] * Wv[(size_t)m*MODEL + jj];
        ho[jj] = s;
    }
    __syncthreads();
    float vals[4]; float ps = 0.f, pq = 0.f;
    for (int q = 0; q < 4; q++) {
        int m = q*256 + tid;
        const float* wr = fcW + (size_t)m * MODEL;
        float s = fcB[m];
        for (int j = 0; j < MODEL; j++) s += ho[j] * wr[j];
        vals[q] = s; ps += s; pq += s*s;
    }
    redS[tid] = ps; redQ[tid] = pq;
    __syncthreads();
    for (int off = 128; off > 0; off >>= 1) {
        if (tid < off) { redS[tid] += redS[tid+off]; redQ[tid] += redQ[tid+off]; }
        __syncthreads();
    }
    float mean = redS[0] * (1.f/MODEL);
    float var  = redQ[0] * (1.f/MODEL) - mean*mean;
    float rs = rsqrtf(var + LN_EPS);
    for (int q = 0; q < 4; q++) {
        int m = q*256 + tid;
        out[(size_t)b*MODEL + m] = (vals[q] - mean) * rs * gamma[m] + beta[m];
    }
}

// ---------------------------------------------------------------------------
extern "C" void kernel_launch(void* const* d_in, const int* in_sizes, int n_in,
                              void* d_out, int out_size, void* d_ws, size_t ws_size,
                              hipStream_t stream)
{
    const float* X   = (const float*)d_in[0];
    const float* Wq  = (const float*)d_in[1];
    const float* Wk  = (const float*)d_in[2];
    const float* Wv  = (const float*)d_in[3];
    const float* fcW = (const float*)d_in[4];
    const float* fcB = (const float*)d_in[5];
    const float* gam = (const float*)d_in[6];
    const float* bet = (const float*)d_in[7];
    const int*  dIdx = (const int*)d_in[8];
    float* ws  = (float*)d_ws;
    float* out = (float*)d_out;
    const int Smax = in_sizes[0] / (BATCH * MODEL);

    const size_t smemC = (size_t)(SUB*ROWP + HEADS*ROWP + 8*256 + SUB*HEADS*2 + HEADS)
                       * sizeof(float);   // ~139 KB (CDNA5 WGP: 320 KB LDS)
    hipFuncSetAttribute((const void*)attn_kernel,
                        hipFuncAttributeMaxDynamicSharedMemorySize, (int)smemC);

    qk_kernel<<<BATCH, 256, 0, stream>>>(X, Wq, Wk, dIdx, ws, Smax);
    attn_kernel<<<BATCH*NSPLIT, 256, smemC, stream>>>(X, dIdx, ws, Smax);
    combine_kernel<<<BATCH*HEADS, 256, 0, stream>>>(ws);
    out_kernel<<<BATCH, 256, 0, stream>>>(Wv, fcW, fcB, gam, bet, ws + OFF_CTX, out);
}